// HDSpatialBlock_82497731821686
// MI455X (gfx1250) — compile-verified
//
#include <hip/hip_runtime.h>
#include <hip/hip_bf16.h>

// ---------------------------------------------------------------------------
// HDSpatialBlock on MI455X (gfx1250, wave32, WMMA + TDM)
//
// out = (y + hd@S) @ Wout  ==  y @ Wout + hd @ (S @ Wout)
// -> precompute W2 = S@Wout, avoiding the 4096x4096x4096 skip GEMM
//    (total dense work ~140 GFLOP instead of ~380).
//
// GEMMs: exact-fp32 V_WMMA_F32_16X16X4_F32, one wave per 64x64 C region
// (4x4 register-blocked WMMA tiles). A/B panels are staged into LDS by the
// Tensor Data Mover (tensor_load_to_lds, TENSORcnt double-buffering), which
// turns the stride-N B-matrix access into a 2D tile DMA.
// Scan: register-resident recurrence (16 fp32 states/thread, 8192 threads),
// B_t/C_t broadcast staged through LDS in 64-step chunks.
// ---------------------------------------------------------------------------

typedef __attribute__((ext_vector_type(2))) float        v2f;
typedef __attribute__((ext_vector_type(8))) float        v8f;
typedef __attribute__((ext_vector_type(4))) unsigned int v4u;
typedef __attribute__((ext_vector_type(8))) int          v8i;
typedef __attribute__((ext_vector_type(4))) int          v4i;

static constexpr int Bn   = 2;
static constexpr int Ln   = 2048;
static constexpr int DIN  = 1024;
static constexpr int HDn  = 4096;
static constexpr int NST  = 16;
static constexpr int ROWS = Bn * Ln;   // 4096

// ---------------------------------------------------------------------------
// TDM: DMA a 2D fp32 tile (tile_d1 rows x tile_d0 elems, row stride stride_d0
// elems) from global into LDS at byte offset lds_byte. D# packing per CDNA5
// ISA 8.3/8.4: group0 = count|lds_addr|global_addr|type=2,
// group1 = data_size=4B | tensor dims | tile dims | dim0 stride. Groups 2/3
// and the trailing group are zero (2D tile: tile_dim2 = 0 -> unused).
// ---------------------------------------------------------------------------
__device__ __forceinline__ void tdm_load_tile_f32(
    unsigned lds_byte, const float* gptr,
    unsigned tensor_d0, unsigned tensor_d1, unsigned stride_d0,
    unsigned tile_d0, unsigned tile_d1)
{
    unsigned long long ga = (unsigned long long)(size_t)gptr;
    v4u g0;
    g0[0] = 1u;                                               // count=1, no gather
    g0[1] = lds_byte;                                         // lds_addr
    g0[2] = (unsigned)(ga & 0xFFFFFFFFu);                     // global_addr[31:0]
    g0[3] = ((unsigned)(ga >> 32) & 0x01FFFFFFu) | (2u << 30);// addr[56:32] | type=2

    v8i g1;
    g1[0] = (int)(2u << 16);                                  // data_size = 4B
    g1[1] = (int)((tensor_d0 & 0xFFFFu) << 16);               // dim0[15:0] (abar=0)
    g1[2] = (int)((tensor_d0 >> 16) | ((tensor_d1 & 0xFFFFu) << 16));
    g1[3] = (int)((tensor_d1 >> 16) | (tile_d0 << 16));
    g1[4] = (int)tile_d1;                                     // tile_dim2 = 0
    g1[5] = (int)stride_d0;                                   // dim0 stride [31:0]
    g1[6] = 0;                                                // stride[47:32]=0, dim1_stride=0
    g1[7] = 0;

    v4i gz4 = {0, 0, 0, 0};
    v8i gz8 = {0, 0, 0, 0, 0, 0, 0, 0};
    __builtin_amdgcn_tensor_load_to_lds(g0, g1, gz4, gz4, gz8, 0);
}

// ---------------------------------------------------------------------------
// fp32 WMMA GEMM:  C[M,N] (+)= A[M,K] @ W[K,N], all row-major.
// One wave per 64x64 C region: acc[4][4] 16x16 tiles, K chunks of 32 staged
// into LDS by the TDM (double-buffered on TENSORcnt).
//
// Fragment layouts (ISA 7.12.2, wave32):
//   A (16x4):  m = lane&15, half = lane>>4; comp j -> A[m][k0 + 2*half + j]
//   B (4x16):  n = lane&15, half = lane>>4; comp j -> W[k0 + 2*half + j][n]
//   C/D 16x16: comp j -> row (j + 8*half), col n = lane&15
// ---------------------------------------------------------------------------
static constexpr int KC = 32;  // K chunk per TDM stage

template <bool ACC>
__global__ __launch_bounds__(32) void gemm_f32_wmma(
    float* __restrict__ C, const float* __restrict__ A,
    const float* __restrict__ W, int M, int N, int K)
{
    __shared__ float sA[2][64 * KC];   // [m][k] : m*KC + k
    __shared__ float sB[2][KC * 64];   // [k][n] : k*64 + n

    const int lane = threadIdx.x;
    const int mn   = lane & 15;        // A-row / B-col within a 16x16 tile
    const int kh   = lane >> 4;        // K half (0/1)
    const int tn   = blockIdx.x * 64;
    const int tm   = blockIdx.y * 64;

    const unsigned ldsA = (unsigned)(size_t)(&sA[0][0]);
    const unsigned ldsB = (unsigned)(size_t)(&sB[0][0]);

    v8f acc[4][4] = {};

    auto issue = [&](int chunk, int buf) {
        const int k0 = chunk * KC;
        // A panel: 64 rows (M) x KC cols, row stride K
        tdm_load_tile_f32(ldsA + (unsigned)buf * sizeof(float) * 64 * KC,
                          A + (size_t)tm * K + k0,
                          (unsigned)K, (unsigned)M, (unsigned)K, KC, 64);
        // W panel: KC rows (K) x 64 cols, row stride N
        tdm_load_tile_f32(ldsB + (unsigned)buf * sizeof(float) * KC * 64,
                          W + (size_t)k0 * N + tn,
                          (unsigned)N, (unsigned)K, (unsigned)N, 64, KC);
    };

    const int nch = K / KC;
    issue(0, 0);

    for (int c = 0; c < nch; ++c) {
        const int buf = c & 1;
        // TDM completes in-order per wave: after issuing chunk c+1's 2 loads,
        // tensorcnt<=2 guarantees chunk c's 2 loads are done.
        if (c + 1 < nch) {
            issue(c + 1, buf ^ 1);
            __builtin_amdgcn_s_wait_tensorcnt(2);
        } else {
            __builtin_amdgcn_s_wait_tensorcnt(0);
        }

        const float* __restrict__ pA = &sA[buf][0];
        const float* __restrict__ pB = &sB[buf][0];

#pragma unroll 4
        for (int k4 = 0; k4 < KC; k4 += 4) {
            const int kk = k4 + 2 * kh;
            v2f af[4], bf[4];
#pragma unroll
            for (int mt = 0; mt < 4; ++mt)
                af[mt] = *(const v2f*)(pA + (mt * 16 + mn) * KC + kk);
#pragma unroll
            for (int nt = 0; nt < 4; ++nt) {
                bf[nt].x = pB[kk * 64 + nt * 16 + mn];
                bf[nt].y = pB[(kk + 1) * 64 + nt * 16 + mn];
            }
#pragma unroll
            for (int mt = 0; mt < 4; ++mt)
#pragma unroll
                for (int nt = 0; nt < 4; ++nt)
                    acc[mt][nt] = __builtin_amdgcn_wmma_f32_16x16x4_f32(
                        false, af[mt], false, bf[nt], (short)0, acc[mt][nt],
                        false, false);
        }
    }

#pragma unroll
    for (int mt = 0; mt < 4; ++mt) {
#pragma unroll
        for (int nt = 0; nt < 4; ++nt) {
            float* __restrict__ Cb =
                C + (size_t)(tm + mt * 16) * N + tn + nt * 16 + mn;
#pragma unroll
            for (int j = 0; j < 8; ++j) {
                const size_t r = (size_t)(j + 8 * kh) * N;
                float v = acc[mt][nt][j];
                if (ACC) v += Cb[r];
                Cb[r] = v;
            }
        }
    }
}

// ---------------------------------------------------------------------------
// Fused B/C projection: Bseq/Cseq[ROWS,16] = (hd * cos(l * phase)) @ {b,c}_proj
// One wave per 16-row tile; Floquet modulation applied to the A fragment, one
// A load feeds two WMMA accumulators. (0.5 GFLOP each - not worth TDM.)
// ---------------------------------------------------------------------------
__global__ __launch_bounds__(32) void bc_proj_wmma(
    float* __restrict__ Bseq, float* __restrict__ Cseq,
    const float* __restrict__ hd, const float* __restrict__ b_proj,
    const float* __restrict__ c_proj, const float* __restrict__ phases)
{
    const int lane = threadIdx.x;
    const int mn   = lane & 15;
    const int kh   = lane >> 4;
    const int tm   = blockIdx.x * 16;
    const int row  = tm + mn;
    const float lf = (float)(row & (Ln - 1));          // l = row % L  (per lane)

    v8f accB = {}, accC = {};
    const float* __restrict__ Arow = hd + (size_t)row * HDn;

#pragma unroll 2
    for (int k0 = 0; k0 < HDn; k0 += 4) {
        const int kk = k0 + 2 * kh;
        v2f a = *(const v2f*)(Arow + kk);
        v2f p = *(const v2f*)(phases + kk);
        a.x *= __cosf(lf * p.x);
        a.y *= __cosf(lf * p.y);
        v2f bb, cc;
        bb.x = b_proj[(size_t)kk * NST + mn];
        bb.y = b_proj[(size_t)(kk + 1) * NST + mn];
        cc.x = c_proj[(size_t)kk * NST + mn];
        cc.y = c_proj[(size_t)(kk + 1) * NST + mn];
        accB = __builtin_amdgcn_wmma_f32_16x16x4_f32(
            false, a, false, bb, (short)0, accB, false, false);
        accC = __builtin_amdgcn_wmma_f32_16x16x4_f32(
            false, a, false, cc, (short)0, accC, false, false);
    }

#pragma unroll
    for (int j = 0; j < 8; ++j) {
        const size_t r = (size_t)(tm + j + 8 * kh) * NST + mn;
        Bseq[r] = accB[j];
        Cseq[r] = accC[j];
    }
}

// ---------------------------------------------------------------------------
// Sequential SSM scan. One thread per (b,d) channel: 16 fp32 states in
// registers, 2048 steps. B_t/C_t (shared by all 4096 threads of one batch)
// staged through LDS in 64-step chunks. x_t recomputed from hd with the cos
// modulation on the fly (saves a 64MB hd_enc buffer).
// ---------------------------------------------------------------------------
static constexpr int SCAN_CH = 64;

__global__ __launch_bounds__(256) void ssm_scan(
    float* __restrict__ y, const float* __restrict__ hd,
    const float* __restrict__ Bseq, const float* __restrict__ Cseq,
    const float* __restrict__ a_log, const float* __restrict__ dt_proj,
    const float* __restrict__ phases)
{
    __shared__ float sB[SCAN_CH * NST];
    __shared__ float sC[SCAN_CH * NST];

    const int tid = blockIdx.x * 256 + threadIdx.x;     // 0..B*HD-1
    const int b   = tid / HDn;                          // whole block shares b
    const int d   = tid % HDn;

    const float dtv = log1pf(__expf(dt_proj[d]));       // softplus
    const float ph  = phases[d];

    float dec[NST], h[NST];
#pragma unroll
    for (int n = 0; n < NST; ++n) {
        dec[n] = __expf(-dtv * __expf(a_log[n]));       // exp(dt*A), A=-exp(a_log)
        h[n]   = 0.0f;
    }

    const float* __restrict__ xb = hd + (size_t)b * Ln * HDn + d;
    float* __restrict__       yb = y  + (size_t)b * Ln * HDn + d;
    const float* __restrict__ Bb = Bseq + (size_t)b * Ln * NST;
    const float* __restrict__ Cb = Cseq + (size_t)b * Ln * NST;

    for (int t0 = 0; t0 < Ln; t0 += SCAN_CH) {
        __syncthreads();
        for (int i = threadIdx.x; i < SCAN_CH * NST; i += 256) {
            sB[i] = Bb[t0 * NST + i];
            sC[i] = Cb[t0 * NST + i];
        }
        __syncthreads();

        for (int tt = 0; tt < SCAN_CH; ++tt) {
            const int t = t0 + tt;
            const float xv = xb[(size_t)t * HDn] * __cosf((float)t * ph);
            const float u  = dtv * xv;
            float yv = 0.0f;
#pragma unroll
            for (int n = 0; n < NST; ++n) {
                h[n] = dec[n] * h[n] + u * sB[tt * NST + n];
                yv  += h[n] * sC[tt * NST + n];
            }
            yb[(size_t)t * HDn] = yv;
        }
    }
}

// ---------------------------------------------------------------------------
// Launch
// ---------------------------------------------------------------------------
extern "C" void kernel_launch(void* const* d_in, const int* in_sizes, int n_in,
                              void* d_out, int out_size, void* d_ws, size_t ws_size,
                              hipStream_t stream) {
    const float* x       = (const float*)d_in[0];   // [B,L,DIN]
    const float* Win     = (const float*)d_in[1];   // [DIN,HD]
    const float* Wout    = (const float*)d_in[2];   // [HD,DIN]
    const float* a_log   = (const float*)d_in[3];   // [N]
    const float* b_proj  = (const float*)d_in[4];   // [HD,N]
    const float* c_proj  = (const float*)d_in[5];   // [HD,N]
    const float* dt_proj = (const float*)d_in[6];   // [HD]
    const float* Sproj   = (const float*)d_in[7];   // [HD,HD]
    const float* phases  = (const float*)d_in[8];   // [HD]
    float* out = (float*)d_out;                     // [B,L,DIN]

    // Workspace (fp32): hd[4096,4096] | y[4096,4096] | W2[4096,1024]
    //                   | Bseq[4096,16] | Cseq[4096,16]   (~151.5 MB)
    float* hd   = (float*)d_ws;
    float* yws  = hd   + (size_t)ROWS * HDn;
    float* W2   = yws  + (size_t)ROWS * HDn;
    float* Bseq = W2   + (size_t)HDn * DIN;
    float* Cseq = Bseq + (size_t)ROWS * NST;

    const dim3 wave(32);

    // 1) hd = x @ Win                       [4096,1024]x[1024,4096]
    gemm_f32_wmma<false><<<dim3(HDn / 64, ROWS / 64), wave, 0, stream>>>(
        hd, x, Win, ROWS, HDn, DIN);

    // 2) Bseq/Cseq = enc(hd) @ {b,c}_proj   [4096,4096]x[4096,16] (fused cos)
    bc_proj_wmma<<<dim3(ROWS / 16), wave, 0, stream>>>(
        Bseq, Cseq, hd, b_proj, c_proj, phases);

    // 3) W2 = S @ Wout                      [4096,4096]x[4096,1024]
    gemm_f32_wmma<false><<<dim3(DIN / 64, HDn / 64), wave, 0, stream>>>(
        W2, Sproj, Wout, HDn, DIN, HDn);

    // 4) y = scan(enc(hd), Bseq, Cseq)      sequential over L
    ssm_scan<<<dim3(Bn * HDn / 256), dim3(256), 0, stream>>>(
        yws, hd, Bseq, Cseq, a_log, dt_proj, phases);

    // 5) out = y @ Wout                     [4096,4096]x[4096,1024]
    gemm_f32_wmma<false><<<dim3(DIN / 64, ROWS / 64), wave, 0, stream>>>(
        out, yws, Wout, ROWS, DIN, HDn);

    // 6) out += hd @ W2                     [4096,4096]x[4096,1024]
    gemm_f32_wmma<true><<<dim3(DIN / 64, ROWS / 64), wave, 0, stream>>>(
        out, hd, W2, ROWS, DIN, HDn);
}